// MoEGate_15015205667494
// MI455X (gfx1250) — compile-verified
//
#include <hip/hip_runtime.h>
#include <hip/hip_bf16.h>

typedef float v2f __attribute__((ext_vector_type(2)));
typedef float v8f __attribute__((ext_vector_type(8)));

#define TOKENS   16384      // B*S = 4*4096
#define HDIM     4096
#define NEXP     64
#define NBATCH   4
#define SEQ      4096
#define TOPK     8

// ---------------------------------------------------------------------------
// Kernel 0: zero the per-(b,e) accumulators (512 floats) every call
// ---------------------------------------------------------------------------
__global__ void zero_accum_kernel(float* __restrict__ p, int n) {
    int i = blockIdx.x * blockDim.x + threadIdx.x;
    if (i < n) p[i] = 0.0f;
}

// ---------------------------------------------------------------------------
// Kernel 1: router GEMM, logits[T][64] = x[T][H] * W[64][H]^T  (fp32 WMMA)
// One wave -> 16 tokens x all 64 experts (4 accumulators of 16x16).
// 8 waves per block, grid = T / 128.
// ---------------------------------------------------------------------------
__global__ __launch_bounds__(256) void router_gemm_kernel(
    const float* __restrict__ x, const float* __restrict__ w,
    float* __restrict__ logits)
{
    const int lane = threadIdx.x & 31;
    const int wave = threadIdx.x >> 5;
    const int m  = lane & 15;       // M / N index within tile
    const int hi = lane >> 4;       // hi half-wave -> K offset +2
    const long token_base = ((long)blockIdx.x * 8 + wave) * 16;

    // A fragment pointer: row = token, per-lane K base = 2*hi (b64 loads)
    const v2f* __restrict__ ap =
        (const v2f*)(x + (token_base + m) * (long)HDIM + 2 * hi);
    // B fragment pointers, one per 16-expert group (weight is row-major [E][H])
    const v2f* __restrict__ bp0 = (const v2f*)(w + (long)( 0 + m) * HDIM + 2 * hi);
    const v2f* __restrict__ bp1 = (const v2f*)(w + (long)(16 + m) * HDIM + 2 * hi);
    const v2f* __restrict__ bp2 = (const v2f*)(w + (long)(32 + m) * HDIM + 2 * hi);
    const v2f* __restrict__ bp3 = (const v2f*)(w + (long)(48 + m) * HDIM + 2 * hi);

    v8f acc0 = {}, acc1 = {}, acc2 = {}, acc3 = {};

    // K loop: 4 per WMMA, unrolled x2 (K=8 per iteration)
    for (int k = 0; k < HDIM; k += 8) {
        const int i0 = k >> 1;       // v2f index: element k + 2*hi
        const int i1 = i0 + 2;       // element k + 4 + 2*hi
        v2f a0 = ap[i0], a1 = ap[i1];
        v2f b00 = bp0[i0], b10 = bp1[i0], b20 = bp2[i0], b30 = bp3[i0];
        v2f b01 = bp0[i1], b11 = bp1[i1], b21 = bp2[i1], b31 = bp3[i1];

        acc0 = __builtin_amdgcn_wmma_f32_16x16x4_f32(false, a0, false, b00,
                                                     (short)0, acc0, false, false);
        acc1 = __builtin_amdgcn_wmma_f32_16x16x4_f32(false, a0, false, b10,
                                                     (short)0, acc1, false, false);
        acc2 = __builtin_amdgcn_wmma_f32_16x16x4_f32(false, a0, false, b20,
                                                     (short)0, acc2, false, false);
        acc3 = __builtin_amdgcn_wmma_f32_16x16x4_f32(false, a0, false, b30,
                                                     (short)0, acc3, false, false);
        acc0 = __builtin_amdgcn_wmma_f32_16x16x4_f32(false, a1, false, b01,
                                                     (short)0, acc0, false, false);
        acc1 = __builtin_amdgcn_wmma_f32_16x16x4_f32(false, a1, false, b11,
                                                     (short)0, acc1, false, false);
        acc2 = __builtin_amdgcn_wmma_f32_16x16x4_f32(false, a1, false, b21,
                                                     (short)0, acc2, false, false);
        acc3 = __builtin_amdgcn_wmma_f32_16x16x4_f32(false, a1, false, b31,
                                                     (short)0, acc3, false, false);
    }

    // D layout: VGPR r -> token row (r + 8*hi), column = lane&15 within group
    float* __restrict__ lrow = logits + (token_base + 8 * hi) * NEXP + m;
#pragma unroll
    for (int r = 0; r < 8; ++r) {
        lrow[(long)r * NEXP +  0] = acc0[r];
        lrow[(long)r * NEXP + 16] = acc1[r];
        lrow[(long)r * NEXP + 32] = acc2[r];
        lrow[(long)r * NEXP + 48] = acc3[r];
    }
}

// ---------------------------------------------------------------------------
// Kernel 2: per-token softmax(64) + top-8 + renormalize; accumulate aux stats.
// One wave per token (2 experts per lane), 8 waves per block.
// ---------------------------------------------------------------------------
__global__ __launch_bounds__(256) void softmax_topk_kernel(
    const float* __restrict__ logits,
    int* __restrict__ out_idx, float* __restrict__ out_w,
    float* __restrict__ gscore, float* __restrict__ gcnt)
{
    __shared__ float sc[NEXP];      // sum of softmax scores per expert (block)
    __shared__ float scnt[NEXP];    // top-k selection counts per expert (block)
    const int tid = threadIdx.x;
    if (tid < NEXP) { sc[tid] = 0.0f; scnt[tid] = 0.0f; }
    __syncthreads();

    const int lane = tid & 31;
    const int wave = tid >> 5;
    const long t = (long)blockIdx.x * 8 + wave;
    const int e0 = lane * 2;

    float2 v = *((const float2*)(logits + t * NEXP) + lane);

    // softmax max
    float mx = fmaxf(v.x, v.y);
#pragma unroll
    for (int off = 16; off > 0; off >>= 1)
        mx = fmaxf(mx, __shfl_xor(mx, off, 32));
    float p0 = __expf(v.x - mx);
    float p1 = __expf(v.y - mx);
    float s = p0 + p1;
#pragma unroll
    for (int off = 16; off > 0; off >>= 1)
        s += __shfl_xor(s, off, 32);
    const float inv = 1.0f / s;
    p0 *= inv; p1 *= inv;

    // accumulate full softmax scores for mean_scores (aux loss)
    atomicAdd(&sc[e0 + 0], p0);
    atomicAdd(&sc[e0 + 1], p1);

    // iterative top-8 with lower-index tie-break
    float q0 = p0, q1 = p1;
    float myv = 0.0f; int myi = 0;
    float ssum = 0.0f;
#pragma unroll
    for (int it = 0; it < TOPK; ++it) {
        float bv; int bi;
        if (q0 >= q1) { bv = q0; bi = e0; } else { bv = q1; bi = e0 + 1; }
#pragma unroll
        for (int off = 16; off > 0; off >>= 1) {
            float ov = __shfl_xor(bv, off, 32);
            int   oi = __shfl_xor(bi, off, 32);
            if (ov > bv || (ov == bv && oi < bi)) { bv = ov; bi = oi; }
        }
        ssum += bv;
        if (lane == it) { myv = bv; myi = bi; }
        if (bi == e0)     q0 = -1e30f;
        if (bi == e0 + 1) q1 = -1e30f;
        if (lane == 0) atomicAdd(&scnt[bi], 1.0f);
    }

    if (lane < TOPK) {
        out_idx[t * TOPK + lane] = myi;
        out_w[t * TOPK + lane]   = myv / (ssum + 1e-20f);
    }

    __syncthreads();
    // flush block accumulators: all 8 tokens of this block share one batch b
    const int b = (int)(blockIdx.x / (SEQ / 8));   // 512 blocks per batch
    if (tid < NEXP) {
        atomicAdd(&gscore[b * NEXP + tid], sc[tid]);
        atomicAdd(&gcnt[b * NEXP + tid], scnt[tid]);
    }
}

// ---------------------------------------------------------------------------
// Kernel 3: aux_loss = ALPHA * mean_b( sum_e( ce[b][e] * mean_scores[b][e] ) )
//   ce = cnt * E/(S*K) = cnt/512 ; mean_scores = score_sum / S
// ---------------------------------------------------------------------------
__global__ __launch_bounds__(256) void aux_loss_kernel(
    const float* __restrict__ gscore, const float* __restrict__ gcnt,
    float* __restrict__ out_aux)
{
    __shared__ float red[NBATCH * NEXP];
    const int i = threadIdx.x;  // 256 = NBATCH*NEXP
    float vv = (gcnt[i] * (float)NEXP / (float)(SEQ * TOPK)) *
               (gscore[i] * (1.0f / (float)SEQ));
    red[i] = vv;
    __syncthreads();
#pragma unroll
    for (int off = 128; off > 0; off >>= 1) {
        if (i < off) red[i] += red[i + off];
        __syncthreads();
    }
    if (i == 0) *out_aux = red[0] * (0.001f / (float)NBATCH);
}

// ---------------------------------------------------------------------------
extern "C" void kernel_launch(void* const* d_in, const int* in_sizes, int n_in,
                              void* d_out, int out_size, void* d_ws, size_t ws_size,
                              hipStream_t stream) {
    const float* x = (const float*)d_in[0];   // (4, 4096, 4096) fp32
    const float* w = (const float*)d_in[1];   // (64, 4096) fp32

    float* logits = (float*)d_ws;                         // T*64 floats (4 MB)
    float* gscore = logits + (long)TOKENS * NEXP;         // 256 floats
    float* gcnt   = gscore + NBATCH * NEXP;               // 256 floats

    float* out     = (float*)d_out;
    int*   out_idx = (int*)out;                           // topk_idx (int32 bits)
    float* out_w   = out + (long)TOKENS * TOPK;           // topk_weight
    float* out_aux = out + 2L * TOKENS * TOPK;            // aux_loss scalar

    zero_accum_kernel<<<1, 512, 0, stream>>>(gscore, 2 * NBATCH * NEXP);
    router_gemm_kernel<<<TOKENS / 128, 256, 0, stream>>>(x, w, logits);
    softmax_topk_kernel<<<TOKENS / 8, 256, 0, stream>>>(logits, out_idx, out_w,
                                                        gscore, gcnt);
    aux_loss_kernel<<<1, NBATCH * NEXP, 0, stream>>>(gscore, gcnt, out_aux);
}